// GraphTransformer_67826123538752
// MI455X (gfx1250) — compile-verified
//
#include <hip/hip_runtime.h>
#include <math.h>

typedef float v2f __attribute__((ext_vector_type(2)));
typedef float v8f __attribute__((ext_vector_type(8)));

// ---------------------------------------------------------------------------
// Projection GEMM: out = x @ W + b for Q/K/V/skip simultaneously.
// Block = 128 threads = 4 waves; wave w computes projection w for a 64-row
// strip of x using V_WMMA_F32_16X16X4_F32 (full fp32 precision).
//
// Weights are staged into LDS pre-swizzled into WMMA B-fragment order:
//   W[k][n] -> Wsh[wave*4096 + frag*64 + lane*2 + j]
//     frag  = (k>>2)*4 + (n>>4)         (K-tile, N-tile)
//     lane  = ((k>>1)&1)*16 + (n&15)    (K-half selector, N-within-tile)
//     j     = k&1
// so each B fragment is one aligned 8-byte ds_load_b64 per lane,
// conflict-free across all 64 LDS banks, landing directly in an even
// VGPR pair (no repacking movs before v_wmma).
// ---------------------------------------------------------------------------
__global__ __launch_bounds__(128) void proj_wmma_kernel(
    const float* __restrict__ x,
    const float* __restrict__ Wq, const float* __restrict__ Wk,
    const float* __restrict__ Wv, const float* __restrict__ Ws,
    const float* __restrict__ bq, const float* __restrict__ bk,
    const float* __restrict__ bv, const float* __restrict__ bs,
    float* __restrict__ oq, float* __restrict__ ok,
    float* __restrict__ ov, float* __restrict__ os,
    int n)
{
    __shared__ __align__(16) float Wsh[4 * 64 * 64]; // 64 KB
    const int tid  = threadIdx.x;
    const int wave = tid >> 5;
    const int lane = tid & 31;

    // Each wave stages its own 64x64 weight matrix into LDS, swizzled
    // into fragment order (coalesced float4 reads from global).
    {
        const float* wsrc = (wave == 0) ? Wq : (wave == 1) ? Wk
                          : (wave == 2) ? Wv : Ws;
        float* WlB = Wsh + wave * 4096;
        for (int i = lane; i < 1024; i += 32) {
            float4 w4 = ((const float4*)wsrc)[i];
            const int k     = i >> 4;          // row 0..63
            const int n0    = (i & 15) * 4;    // col 0,4,...,60
            const int kt    = k >> 2;
            const int kr    = k & 3;
            const int khalf = kr >> 1;
            const int j     = kr & 1;
            const int nt    = n0 >> 4;         // same nt for all 4 cols
            const int m0    = n0 & 15;
            const int base  = (kt * 4 + nt) * 64 + j;
            WlB[base + (khalf * 16 + m0    ) * 2] = w4.x;
            WlB[base + (khalf * 16 + m0 + 1) * 2] = w4.y;
            WlB[base + (khalf * 16 + m0 + 2) * 2] = w4.z;
            WlB[base + (khalf * 16 + m0 + 3) * 2] = w4.w;
        }
    }
    __syncthreads();

    const v2f* Wfrag = (const v2f*)(Wsh + wave * 4096);
    const float* bl  = (wave == 0) ? bq : (wave == 1) ? bk
                     : (wave == 2) ? bv : bs;
    float*       outp= (wave == 0) ? oq : (wave == 1) ? ok
                     : (wave == 2) ? ov : os;

    const int mrow  = lane & 15;   // M (A) / N (B,D) index within tile
    const int khalf = lane >> 4;   // K half selector for A fragments
    const int row_base = blockIdx.x * 64;

    for (int rt = 0; rt < 4; ++rt) {
        const int r0 = row_base + rt * 16;
        if (r0 >= n) break;                       // wave-uniform exit

        v8f acc[4];
        #pragma unroll
        for (int nt = 0; nt < 4; ++nt) {
            v8f z = {0.f,0.f,0.f,0.f,0.f,0.f,0.f,0.f};
            acc[nt] = z;
        }

        int arow = r0 + mrow;
        if (arow >= n) arow = n - 1;              // clamp (no EXEC change)
        const float* xr = x + (size_t)arow * 64;

        #pragma unroll
        for (int kt = 0; kt < 16; ++kt) {
            const int k0 = kt * 4 + khalf * 2;
            v2f a;
            a.x = xr[k0];
            a.y = xr[k0 + 1];
            #pragma unroll
            for (int nt = 0; nt < 4; ++nt) {
                const v2f b = Wfrag[(kt * 4 + nt) * 32 + lane];
                acc[nt] = __builtin_amdgcn_wmma_f32_16x16x4_f32(
                    false, a, false, b, (short)0, acc[nt], false, false);
            }
        }

        #pragma unroll
        for (int nt = 0; nt < 4; ++nt) {
            const float bias = bl[nt * 16 + mrow];
            #pragma unroll
            for (int r = 0; r < 8; ++r) {
                const int row = r0 + khalf * 8 + r;   // C/D VGPR layout
                if (row < n)
                    outp[(size_t)row * 64 + nt * 16 + mrow] = acc[nt][r] + bias;
            }
        }
    }
}

// ---------------------------------------------------------------------------
// Elementwise / edge kernels
// ---------------------------------------------------------------------------
__global__ void zero_kernel(float* __restrict__ p, long long count)
{
    long long i = (long long)blockIdx.x * blockDim.x + threadIdx.x;
    if (i < count) p[i] = 0.f;
}

__global__ void init_nodes(float* __restrict__ m, float* __restrict__ denom, int n)
{
    int i = blockIdx.x * blockDim.x + threadIdx.x;
    if (i < n) { m[i] = -__builtin_huge_valf(); denom[i] = 0.f; }
}

// s[e] = dot(q[dst], k[src]) / sqrt(D)
__global__ void score_kernel(const float* __restrict__ q,
                             const float* __restrict__ k,
                             const long long* __restrict__ src,
                             const long long* __restrict__ dst,
                             float* __restrict__ s, long long E)
{
    long long e = (long long)blockIdx.x * blockDim.x + threadIdx.x;
    if (e >= E) return;
    const int si = (int)src[e], di = (int)dst[e];
    const float4* qa = (const float4*)(q + (size_t)di * 64);
    const float4* ka = (const float4*)(k + (size_t)si * 64);
    float acc = 0.f;
    #pragma unroll
    for (int j = 0; j < 16; ++j) {
        float4 a = qa[j], b = ka[j];
        acc += a.x * b.x + a.y * b.y + a.z * b.z + a.w * b.w;
    }
    s[e] = acc * 0.125f; // 1/sqrt(64)
}

// monotone int/uint mapping trick -> float atomic max for signed values
__device__ __forceinline__ void atomicMaxFloat(float* addr, float v)
{
    if (v >= 0.f) atomicMax((int*)addr, __float_as_int(v));
    else          atomicMin((unsigned int*)addr, __float_as_uint(v));
}

__global__ void max_kernel(const float* __restrict__ s,
                           const long long* __restrict__ dst,
                           float* __restrict__ m, long long E)
{
    long long e = (long long)blockIdx.x * blockDim.x + threadIdx.x;
    if (e >= E) return;
    atomicMaxFloat(&m[(int)dst[e]], s[e]);
}

__global__ void exp_kernel(const float* __restrict__ s,
                           const long long* __restrict__ dst,
                           const float* __restrict__ m,
                           float* __restrict__ ebuf,
                           float* __restrict__ denom, long long E)
{
    long long e = (long long)blockIdx.x * blockDim.x + threadIdx.x;
    if (e >= E) return;
    const int di = (int)dst[e];
    const float v = expf(s[e] - m[di]);
    ebuf[e] = v;
    atomicAdd(&denom[di], v);
}

// hout[dst] += (e/denom[dst]) * v[src]; 64 threads per edge (one per feature)
__global__ void agg_kernel(const float* __restrict__ ebuf,
                           const float* __restrict__ denom,
                           const float* __restrict__ v,
                           const long long* __restrict__ src,
                           const long long* __restrict__ dst,
                           float* __restrict__ hout, long long E)
{
    long long idx = (long long)blockIdx.x * blockDim.x + threadIdx.x;
    long long e = idx >> 6;
    int d = (int)(idx & 63);
    if (e >= E) return;
    const int si = (int)src[e], di = (int)dst[e];
    const float alpha = ebuf[e] / denom[di];
    atomicAdd(&hout[(size_t)di * 64 + d], alpha * v[(size_t)si * 64 + d]);
}

__global__ void add_kernel(float* __restrict__ h,
                           const float* __restrict__ sk, long long count)
{
    long long i = (long long)blockIdx.x * blockDim.x + threadIdx.x;
    if (i < count) h[i] += sk[i];
}

// BatchNorm stats: block b owns feature b; LDS tree reduction over N nodes.
__global__ __launch_bounds__(256) void bn_stats_kernel(
    const float* __restrict__ h, float* __restrict__ mean,
    float* __restrict__ rstd, int n)
{
    const int f = blockIdx.x;       // 0..63
    float s = 0.f, s2 = 0.f;
    for (int i = threadIdx.x; i < n; i += 256) {
        float x = h[(size_t)i * 64 + f];
        s += x; s2 += x * x;
    }
    __shared__ float sh[256], sh2[256];
    sh[threadIdx.x] = s; sh2[threadIdx.x] = s2;
    __syncthreads();
    for (int off = 128; off > 0; off >>= 1) {
        if (threadIdx.x < off) {
            sh[threadIdx.x]  += sh[threadIdx.x + off];
            sh2[threadIdx.x] += sh2[threadIdx.x + off];
        }
        __syncthreads();
    }
    if (threadIdx.x == 0) {
        float mu  = sh[0] / (float)n;
        float var = sh2[0] / (float)n - mu * mu;
        mean[f] = mu;
        rstd[f] = rsqrtf(var + 1e-5f);
    }
}

__global__ void bn_relu_kernel(float* __restrict__ h,
                               const float* __restrict__ mean,
                               const float* __restrict__ rstd,
                               const float* __restrict__ gamma,
                               const float* __restrict__ beta,
                               long long count)
{
    long long i = (long long)blockIdx.x * blockDim.x + threadIdx.x;
    if (i >= count) return;
    const int f = (int)(i & 63);
    float y = gamma[f] * (h[i] - mean[f]) * rstd[f] + beta[f];
    h[i] = y > 0.f ? y : 0.f;
}

// ---------------------------------------------------------------------------
// Host side
// ---------------------------------------------------------------------------
extern "C" void kernel_launch(void* const* d_in, const int* in_sizes, int n_in,
                              void* d_out, int out_size, void* d_ws, size_t ws_size,
                              hipStream_t stream)
{
    const float*     x    = (const float*)d_in[0];
    const long long* ei   = (const long long*)d_in[1];
    const int        N    = in_sizes[0] / 64;
    const long long  E    = in_sizes[1] / 2;
    const long long* src  = ei;
    const long long* dstp = ei + E;
    const float* Wq = (const float*)d_in[2];
    const float* bq = (const float*)d_in[3];
    const float* Wk = (const float*)d_in[4];
    const float* bk = (const float*)d_in[5];
    const float* Wv = (const float*)d_in[6];
    const float* bv = (const float*)d_in[7];
    const float* Ws = (const float*)d_in[8];
    const float* bs = (const float*)d_in[9];
    const float* gamma = (const float*)d_in[10];
    const float* beta  = (const float*)d_in[11];

    float* ws = (float*)d_ws;
    const size_t nd = (size_t)N * 64;
    float* q    = ws; ws += nd;
    float* kb   = ws; ws += nd;
    float* vb   = ws; ws += nd;
    float* sk   = ws; ws += nd;
    float* hA   = ws; ws += nd;
    float* hB   = ws; ws += nd;
    float* sbuf = ws; ws += E;
    float* ebuf = ws; ws += E;
    float* mbuf = ws; ws += N;
    float* dbuf = ws; ws += N;
    float* mean = ws; ws += 64;
    float* rstd = ws; ws += 64;

    const int gN    = (N + 255) / 256;
    const int gND   = (int)((nd + 255) / 256);
    const int gE    = (int)((E + 255) / 256);
    const int gE64  = (int)((E * 64 + 255) / 256);
    const int gProj = (N + 63) / 64;

    for (int l = 0; l < 3; ++l) {
        const float* hin  = (l == 0) ? x : ((l == 1) ? hA : hB);
        float*       hout = (l == 2) ? (float*)d_out : ((l == 0) ? hA : hB);

        proj_wmma_kernel<<<gProj, 128, 0, stream>>>(
            hin,
            Wq + (size_t)l * 4096, Wk + (size_t)l * 4096,
            Wv + (size_t)l * 4096, Ws + (size_t)l * 4096,
            bq + (size_t)l * 64,   bk + (size_t)l * 64,
            bv + (size_t)l * 64,   bs + (size_t)l * 64,
            q, kb, vb, sk, N);

        zero_kernel<<<gND, 256, 0, stream>>>(hout, (long long)nd);
        init_nodes<<<gN, 256, 0, stream>>>(mbuf, dbuf, N);
        score_kernel<<<gE, 256, 0, stream>>>(q, kb, src, dstp, sbuf, E);
        max_kernel<<<gE, 256, 0, stream>>>(sbuf, dstp, mbuf, E);
        exp_kernel<<<gE, 256, 0, stream>>>(sbuf, dstp, mbuf, ebuf, dbuf, E);
        agg_kernel<<<gE64, 256, 0, stream>>>(ebuf, dbuf, vb, src, dstp, hout, E);
        add_kernel<<<gND, 256, 0, stream>>>(hout, sk, (long long)nd);

        if (l < 2) {
            bn_stats_kernel<<<64, 256, 0, stream>>>(hout, mean, rstd, N);
            bn_relu_kernel<<<gND, 256, 0, stream>>>(hout, mean, rstd,
                gamma + (size_t)l * 64, beta + (size_t)l * 64, (long long)nd);
        }
    }
}